// SNNLayer_57930518888891
// MI455X (gfx1250) — compile-verified
//
#include <hip/hip_runtime.h>

typedef __attribute__((ext_vector_type(2))) float v2f;
typedef __attribute__((ext_vector_type(8))) float v8f;

#define B_ 16
#define C_ 1024
#define T_ 2048
#define N_ 1024

#define CSPLIT 4           // phase-1 c-dimension split (occupancy)
#define CCHUNK (C_ / CSPLIT)
#define OSPLIT 16          // phase-2 o-dimension split
#define OCHUNK (N_ / OSPLIT)

// ---------------------------------------------------------------------------
// Phase 1: xs_part[s][b][t] = sum_{c in chunk s} x[b,c,t]
// Matrix-pipe reduction: D = Ones(16x4) * B(4x16) + C via v_wmma_f32_16x16x4_f32.
// Wave owns (b, 16-wide t-tile, c-chunk). Lane L supplies B column n = L%16;
// the four (vgpr, lane-half) slots carry four distinct c-rows, so sum_k B[k,n]
// is exact regardless of the internal k ordering. Two accumulator chains.
// 8192 waves => ~31 MB of loads in flight (covers 23.3 TB/s * ~700 ns).
// ---------------------------------------------------------------------------
__global__ __launch_bounds__(256) void snn_colsum_x(const float* __restrict__ x,
                                                    float* __restrict__ xs_part) {
  const int gtid  = blockIdx.x * blockDim.x + threadIdx.x;
  const int wave  = gtid >> 5;
  const int lane  = gtid & 31;
  const int split = wave & 3;
  const int tile  = (wave >> 2) & 127;
  const int b     = wave >> 9;
  const int t0    = tile * 16;
  const int half  = lane >> 4;    // 0 for lanes 0-15, 1 for lanes 16-31
  const int tt    = lane & 15;    // N column index
  const int cbase = split * CCHUNK;

  const float* xb = x + (size_t)b * C_ * T_ + (size_t)(t0 + tt);

  v2f a;  a.x = 1.0f; a.y = 1.0f;   // all-ones A matrix (16x4)
  v8f acc0 = {0.f,0.f,0.f,0.f,0.f,0.f,0.f,0.f};
  v8f acc1 = {0.f,0.f,0.f,0.f,0.f,0.f,0.f,0.f};

  for (int c0 = cbase; c0 < cbase + CCHUNK; c0 += 8) {
    v2f b0, b1;
    b0.x = xb[(size_t)(c0 + 0 + half) * T_];
    b0.y = xb[(size_t)(c0 + 2 + half) * T_];
    b1.x = xb[(size_t)(c0 + 4 + half) * T_];
    b1.y = xb[(size_t)(c0 + 6 + half) * T_];
    acc0 = __builtin_amdgcn_wmma_f32_16x16x4_f32(false, a, false, b0,
                                                 (short)0, acc0, false, false);
    acc1 = __builtin_amdgcn_wmma_f32_16x16x4_f32(false, a, false, b1,
                                                 (short)0, acc1, false, false);
  }

  // All D rows are identical (ones A); acc[0] on lane L holds the partial sum
  // for t0 + L%16.
  float r = acc0[0] + acc1[0];
  if (lane < 16) {
    xs_part[((size_t)split * B_ + b) * T_ + t0 + tt] = r;
  }
}

// ---------------------------------------------------------------------------
// Phase 2a: ws_part[k][n] = sum_{o in chunk k} weight[o,n]
// 16384 threads (512 waves) so the o-reduction is latency-parallel, not a
// 1024-deep serial chain. Coalesced across n.
// ---------------------------------------------------------------------------
__global__ __launch_bounds__(256) void snn_colsum_w_part(const float* __restrict__ w,
                                                         float* __restrict__ ws_part) {
  const int n      = (blockIdx.x & 3) * 256 + threadIdx.x;   // 0..1023
  const int ochunk = blockIdx.x >> 2;                        // 0..15
  const int obase  = ochunk * OCHUNK;
  float s = 0.0f;
  for (int o = obase; o < obase + OCHUNK; ++o)
    s += w[(size_t)o * C_ + n];
  ws_part[(size_t)ochunk * C_ + n] = s;
}

// Phase 2b: ws[n] = sum_k ws_part[k][n]
__global__ __launch_bounds__(256) void snn_colsum_w_reduce(const float* __restrict__ ws_part,
                                                           float* __restrict__ ws) {
  const int n = blockIdx.x * blockDim.x + threadIdx.x;
  if (n >= C_) return;
  float s = 0.0f;
#pragma unroll
  for (int k = 0; k < OSPLIT; ++k)
    s += ws_part[(size_t)k * C_ + n];
  ws[n] = s;
}

// ---------------------------------------------------------------------------
// Phase 3: LIF scan. Thread owns one (b,n); xs[b,:] assembled from the 4
// phase-1 partials into LDS (broadcast reads); spikes staged in padded LDS
// (stride 33 -> conflict-free on 64 banks) and written 128B-contiguous/wave.
// ---------------------------------------------------------------------------
#define P3_THREADS 64
#define P3_CT 32

__global__ __launch_bounds__(P3_THREADS) void snn_scan(const float* __restrict__ xs_part,
                                                       const float* __restrict__ ws,
                                                       const float* __restrict__ v0,
                                                       float* __restrict__ out) {
  __shared__ float xsrow[T_];
  __shared__ float stage[P3_THREADS * 33];

  const int tid  = threadIdx.x;
  const int b    = blockIdx.x >> 4;   // N_/P3_THREADS = 16 n-blocks per b
  const int nblk = blockIdx.x & 15;
  const int n    = nblk * P3_THREADS + tid;

  for (int i = tid; i < T_; i += P3_THREADS) {
    const size_t idx = (size_t)b * T_ + i;
    float s = xs_part[idx]
            + xs_part[(size_t)1 * B_ * T_ + idx]
            + xs_part[(size_t)2 * B_ * T_ + idx]
            + xs_part[(size_t)3 * B_ * T_ + idx];
    xsrow[i] = s;
  }
  __syncthreads();

  const float wsn = ws[n];
  float v = v0[n];
  float* obase = out + ((size_t)b * N_ + (size_t)nblk * P3_THREADS) * T_;

  for (int tc = 0; tc < T_; tc += P3_CT) {
#pragma unroll
    for (int j = 0; j < P3_CT; ++j) {
      const float inp = xsrow[tc + j] * wsn;
      v = v + (inp - v) * 0.1f;                   // leaky integrate
      const float s = (v >= 1.0f) ? 1.0f : 0.0f;  // fire
      v = (s > 0.0f) ? 0.0f : v;                  // hard reset
      stage[tid * 33 + j] = s;
    }
    __syncthreads();
    // 64x32 floats -> 32 coalesced wave-wide b32 store rounds (128B/wave)
#pragma unroll
    for (int it = 0; it < P3_CT; ++it) {
      const int l   = it * P3_THREADS + tid;
      const int row = l >> 5;   // which n-row of this chunk
      const int col = l & 31;   // t within chunk
      obase[(size_t)row * T_ + tc + col] = stage[row * 33 + col];
    }
    __syncthreads();
  }
}

// ---------------------------------------------------------------------------
extern "C" void kernel_launch(void* const* d_in, const int* in_sizes, int n_in,
                              void* d_out, int out_size, void* d_ws, size_t ws_size,
                              hipStream_t stream) {
  (void)in_sizes; (void)n_in; (void)out_size; (void)ws_size;
  const float* x  = (const float*)d_in[0];
  const float* w  = (const float*)d_in[1];
  const float* v0 = (const float*)d_in[2];
  float* out = (float*)d_out;

  float* xs_part = (float*)d_ws;                          // CSPLIT*B_*T_ = 131072 floats
  float* ws_part = xs_part + (size_t)CSPLIT * B_ * T_;    // OSPLIT*C_   = 16384 floats
  float* ws      = ws_part + (size_t)OSPLIT * C_;         // C_          = 1024 floats

  // Phase 1: 16 b * 128 tiles * 4 c-chunks = 8192 waves = 262144 threads
  snn_colsum_x<<<1024, 256, 0, stream>>>(x, xs_part);
  // Phase 2a: 16 o-chunks * 1024 n = 64 blocks
  snn_colsum_w_part<<<OSPLIT * (C_ / 256), 256, 0, stream>>>(w, ws_part);
  // Phase 2b: combine 16 partials per n
  snn_colsum_w_reduce<<<(C_ + 255) / 256, 256, 0, stream>>>(ws_part, ws);
  // Phase 3: 16 b * 16 n-blocks = 256 blocks of 64 threads
  snn_scan<<<B_ * (N_ / P3_THREADS), P3_THREADS, 0, stream>>>(xs_part, ws, v0, out);
}